// CANLayer_29592324670274
// MI455X (gfx1250) — compile-verified
//
#include <hip/hip_runtime.h>

typedef __attribute__((ext_vector_type(2))) float v2f;
typedef __attribute__((ext_vector_type(4))) float v4f;
typedef __attribute__((ext_vector_type(8))) float v8f;

#define D 128            // all feature dims are 128
#define KPAIR_STRIDE 288 // dwords per k-pair row in LDS; 288 % 64 == 32 -> half-waves
                         // hit complementary bank halves (conflict-free b64 loads)

// ---------------------------------------------------------------------------
// h = relu(x) @ W      x:[nrows,128] f32, W:[128,128] f32, h:[nrows,128] f32
// Block = 512 threads = 16 waves. W staged once per block into LDS, interleaved
// as sW[(k/2)*288 + col*2 + (k&1)] so each B fragment is ONE ds_load_b64.
// Each wave owns one 16-row strip: 8 accumulators (64 VGPRs) -> no spills.
//
// ISA layouts (CDNA5 ISA, V_WMMA_F32_16X16X4_F32):
//   A 16x4 f32 : lanes 0-15 -> M=0..15 with K0,K1 ; lanes 16-31 -> K2,K3
//   B 4x16 f32 : VGPR0 -> row K0 (lanes 0-15) / K2 (lanes 16-31); VGPR1 -> K1/K3
//   C/D 16x16  : VGPR v -> row M=v (lanes 0-15), M=v+8 (lanes 16-31), N=lane&15
// ---------------------------------------------------------------------------
__global__ void __launch_bounds__(512)
gemm_relu_wmma(const float* __restrict__ x,
               const float* __restrict__ W,
               float* __restrict__ h,
               int nstrips) {
  __shared__ float sW[(D / 2) * KPAIR_STRIDE];   // 64 k-pairs * 288 dwords = 72KB

  const int tid = threadIdx.x;

  // stage W (128x128 f32 = 64KB) into interleaved LDS layout
#pragma unroll
  for (int i = 0; i < 8; ++i) {
    const int chunk = i * 512 + tid;             // 4096 float4 chunks
    const int row = chunk >> 5;                  // k index 0..127
    const int c4  = (chunk & 31) * 4;            // col 0..124 step 4
    v4f w = *(const v4f*)(W + (size_t)row * D + c4);
    float* dst = &sW[(row >> 1) * KPAIR_STRIDE + c4 * 2 + (row & 1)];
    dst[0] = w.x;                                // col c4
    dst[2] = w.y;                                // col c4+1
    dst[4] = w.z;
    dst[6] = w.w;
  }
  __syncthreads();

  const int strip = blockIdx.x * (blockDim.x >> 5) + (tid >> 5);
  if (strip >= nstrips) return;                  // wave-uniform: EXEC stays all-1s

  const int lane = tid & 31;
  const int half = lane >> 4;                    // 0: K0/K1 half, 1: K2/K3 half
  const int l16  = lane & 15;

  const float* xrow = x + (size_t)(strip * 16 + l16) * D;

  v8f acc[8] = {};

  for (int k0 = 0; k0 < D; k0 += 4) {
    // A fragment: relu applied to input activations
    v4f xa = *(const v4f*)(xrow + k0);
    v2f a;
    a.x = fmaxf(half ? xa.z : xa.x, 0.0f);
    a.y = fmaxf(half ? xa.w : xa.y, 0.0f);

    // B fragments: k-pair (k0/2 + half) -> rows (k0+2*half, k0+2*half+1)
    const float* wbase = &sW[((k0 >> 1) + half) * KPAIR_STRIDE + l16 * 2];
#pragma unroll
    for (int t = 0; t < 8; ++t) {
      v2f b = *(const v2f*)(wbase + t * 32);     // one ds_load_b64
      acc[t] = __builtin_amdgcn_wmma_f32_16x16x4_f32(
          false, a, false, b, (short)0, acc[t], false, false);
    }
  }

  const int rbase = strip * 16 + half * 8;
#pragma unroll
  for (int t = 0; t < 8; ++t) {
#pragma unroll
    for (int v = 0; v < 8; ++v) {
      h[(size_t)(rbase + v) * D + t * 16 + l16] = acc[t][v];
    }
  }
}

// ---------------------------------------------------------------------------
// y[rows[e], :] += vals[e] * h[cols[e], :]   (scatter-add, one wave per edge)
// lane handles features [4*lane, 4*lane+3]: one b128 gather + 4 f32 atomics
// ---------------------------------------------------------------------------
__global__ void spmm_scatter_atomic(const int* __restrict__ rows,
                                    const int* __restrict__ cols,
                                    const float* __restrict__ vals,
                                    const float* __restrict__ h,
                                    float* __restrict__ y,
                                    int nnz) {
  const int e = blockIdx.x * (blockDim.x >> 5) + (threadIdx.x >> 5);
  if (e >= nnz) return;
  const int lane = threadIdx.x & 31;

  const int r = rows[e];
  const int c = cols[e];
  const float v = vals[e];

  v4f src = *(const v4f*)(h + (size_t)c * D + 4 * lane);
  float* dst = y + (size_t)r * D + 4 * lane;

  // lowers to global_atomic_add_f32 (no CAS loop)
  (void)__hip_atomic_fetch_add(dst + 0, v * src.x, __ATOMIC_RELAXED, __HIP_MEMORY_SCOPE_AGENT);
  (void)__hip_atomic_fetch_add(dst + 1, v * src.y, __ATOMIC_RELAXED, __HIP_MEMORY_SCOPE_AGENT);
  (void)__hip_atomic_fetch_add(dst + 2, v * src.z, __ATOMIC_RELAXED, __HIP_MEMORY_SCOPE_AGENT);
  (void)__hip_atomic_fetch_add(dst + 3, v * src.w, __ATOMIC_RELAXED, __HIP_MEMORY_SCOPE_AGENT);
}

// ---------------------------------------------------------------------------
// elementwise helpers (all sizes divisible by 4 floats)
// ---------------------------------------------------------------------------
__global__ void fill_zero4(float* __restrict__ p, int n4) {
  const int i = blockIdx.x * blockDim.x + threadIdx.x;
  if (i < n4) {
    v4f z = {0.0f, 0.0f, 0.0f, 0.0f};
    ((v4f*)p)[i] = z;
  }
}

__global__ void relu_copy4(const float* __restrict__ in, float* __restrict__ out, int n4) {
  const int i = blockIdx.x * blockDim.x + threadIdx.x;
  if (i < n4) {
    v4f v = ((const v4f*)in)[i];
    v.x = fmaxf(v.x, 0.0f);
    v.y = fmaxf(v.y, 0.0f);
    v.z = fmaxf(v.z, 0.0f);
    v.w = fmaxf(v.w, 0.0f);
    ((v4f*)out)[i] = v;
  }
}

__global__ void relu_inplace4(float* __restrict__ p, int n4) {
  const int i = blockIdx.x * blockDim.x + threadIdx.x;
  if (i < n4) {
    v4f v = ((v4f*)p)[i];
    v.x = fmaxf(v.x, 0.0f);
    v.y = fmaxf(v.y, 0.0f);
    v.z = fmaxf(v.z, 0.0f);
    v.w = fmaxf(v.w, 0.0f);
    ((v4f*)p)[i] = v;
  }
}

// ---------------------------------------------------------------------------
extern "C" void kernel_launch(void* const* d_in, const int* in_sizes, int n_in,
                              void* d_out, int out_size, void* d_ws, size_t ws_size,
                              hipStream_t stream) {
  constexpr int N0 = 100000, N1 = 150000, N2 = 50000;
  constexpr int NNZ00 = 1600000, NNZ12 = 200000;

  const float* x0       = (const float*)d_in[0];
  const float* x1       = (const float*)d_in[1];
  const int*   n00_rows = (const int*)d_in[2];
  const int*   n00_cols = (const int*)d_in[3];
  const float* n00_vals = (const float*)d_in[4];
  const int*   n12_rows = (const int*)d_in[5];
  const int*   n12_cols = (const int*)d_in[6];
  const float* n12_vals = (const float*)d_in[7];
  const float* W0       = (const float*)d_in[8];
  const float* W12      = (const float*)d_in[9];

  float* out = (float*)d_out;
  float* y0      = out;                                  // [N0,128]
  float* x1_out  = out + (size_t)N0 * D;                 // [N1,128]
  float* y2      = out + (size_t)(N0 + N1) * D;          // [N2,128]

  float* h0 = (float*)d_ws;                              // [N0,128] = 51.2 MB
  float* h1 = h0 + (size_t)N0 * D;                       // [N1,128] = 76.8 MB

  const int threads = 256;                               // 8 waves (wave32)
  const int wavesPerBlock = threads / 32;

  // 1) zero accumulators
  {
    int n4 = N0 * D / 4;
    fill_zero4<<<(n4 + threads - 1) / threads, threads, 0, stream>>>(y0, n4);
    n4 = N2 * D / 4;
    fill_zero4<<<(n4 + threads - 1) / threads, threads, 0, stream>>>(y2, n4);
  }

  // 2) x_1 output = relu(x_1)
  {
    const int n4 = N1 * D / 4;
    relu_copy4<<<(n4 + threads - 1) / threads, threads, 0, stream>>>(x1, x1_out, n4);
  }

  // 3) h0 = relu(x0) @ W0 ; h1 = relu(x1) @ W12   (WMMA f32, W staged in LDS)
  {
    const int gemmThreads = 512;                         // 16 waves per block
    const int gemmWaves = gemmThreads / 32;
    const int strips0 = N0 / 16;                         // 6250
    const int strips1 = N1 / 16;                         // 9375
    gemm_relu_wmma<<<(strips0 + gemmWaves - 1) / gemmWaves, gemmThreads, 0, stream>>>(
        x0, W0, h0, strips0);
    gemm_relu_wmma<<<(strips1 + gemmWaves - 1) / gemmWaves, gemmThreads, 0, stream>>>(
        x1, W12, h1, strips1);
  }

  // 4) scatter-add SpMM (one wave per edge)
  spmm_scatter_atomic<<<(NNZ00 + wavesPerBlock - 1) / wavesPerBlock, threads, 0, stream>>>(
      n00_rows, n00_cols, n00_vals, h0, y0, NNZ00);
  spmm_scatter_atomic<<<(NNZ12 + wavesPerBlock - 1) / wavesPerBlock, threads, 0, stream>>>(
      n12_rows, n12_cols, n12_vals, h1, y2, NNZ12);

  // 5) final relu on y0, y2
  {
    int n4 = N0 * D / 4;
    relu_inplace4<<<(n4 + threads - 1) / threads, threads, 0, stream>>>(y0, n4);
    n4 = N2 * D / 4;
    relu_inplace4<<<(n4 + threads - 1) / threads, threads, 0, stream>>>(y2, n4);
  }
}